// GNNNodeClassifier_31361851195877
// MI455X (gfx1250) — compile-verified
//
#include <hip/hip_runtime.h>
#include <math.h>

typedef __attribute__((ext_vector_type(2))) float v2f;
typedef __attribute__((ext_vector_type(8))) float v8f;

#define GNN_N 50000
#define GNN_E 800000
#define GNN_Q 4096

__device__ __forceinline__ float gelu_tanh(float x) {
    // jax.nn.gelu default (approximate=True)
    float u = 0.7978845608028654f * (x + 0.044715f * x * x * x);
    return 0.5f * x * (1.0f + tanhf(u));
}

// ---------------------------------------------------------------------------
// Fused FFN tile kernel (WMMA fp32 16x16x4), 8 waves / block:
//   O = gelu(X @ W1 + B1) @ W2 + B2
//   mode 0: Y = O
//   mode 1: Y = l2norm(O) + RES            (graph-conv update epilogue)
// X is virtual-concat: K in [0,64) from X0, K in [64,128) from X1.
// Weights are staged once per block into LDS, row-pair interleaved so a
// 4x16 B-fragment is a single ds_load_b64. Each wave owns a 16x64 LDS slice
// used in turn for: A-tile stage (coalesced b128 global loads) -> gelu(H)
// re-layout -> output staging (coalesced b128 global stores).
// Intra-wave LDS ops are in-order (DScnt), so only the weight stage needs a
// block barrier. Tail tiles are clamped (stores guarded) so EXEC is all-ones
// at every WMMA.
// ---------------------------------------------------------------------------
__global__ __launch_bounds__(256)
void ffn_tile_kernel(const float* X0, const float* X1,
                     const float* __restrict__ W1, const float* __restrict__ B1,
                     const float* __restrict__ W2, const float* __restrict__ B2,
                     const float* RES, float* Y, int ntiles, int din, int mode)
{
    __shared__ float wlds[12288];      // W1 pairs [0,8192) + W2 pairs [8192,12288)
    __shared__ float slice[8 * 1024];  // per-wave 16x64 work tile
    __shared__ float nrm[8 * 32];      // per-wave row sum-of-squares partials

    const int tid  = threadIdx.x;
    const int wave = tid >> 5;
    const int lane = tid & 31;
    const int half = lane >> 4;        // K-half selector for A/B frags
    const int lm   = lane & 15;        // row (A) / column-in-tile (B,C)
    const int W2OFF = 8192;

    // ---- stage weights, pair-interleaved: wlds[((k>>1)*64+c)*2 + (k&1)] ----
    const int nw1 = din << 6;
    for (int i = tid; i < nw1; i += 256) {
        int k = i >> 6, c = i & 63;
        wlds[(((k >> 1) << 6) + c) * 2 + (k & 1)] = W1[i];
    }
    for (int i = tid; i < 4096; i += 256) {
        int k = i >> 6, c = i & 63;
        wlds[W2OFF + (((k >> 1) << 6) + c) * 2 + (k & 1)] = W2[i];
    }
    __syncthreads();

    const int t      = blockIdx.x * 8 + wave;
    const bool active = (t < ntiles);
    const int tt     = active ? t : (ntiles - 1);
    const int row0   = tt << 4;
    float* sl = slice + (wave << 10);
    float* nr = nrm + (wave << 5);

    // ---- GEMM1: H = gelu(X @ W1 + B1) -------------------------------------
    v8f acc[4];
    #pragma unroll
    for (int n = 0; n < 4; ++n) acc[n] = (v8f)0.0f;

    const int nhalves = din >> 6;      // 1 or 2
    for (int hph = 0; hph < nhalves; ++hph) {
        const float* Xs = (hph == 0) ? X0 : X1;
        // stage 16x64 A tile with coalesced float4 loads
        #pragma unroll
        for (int i = 0; i < 8; ++i) {
            int j = lane + (i << 5);            // float4 index 0..255
            int r = j >> 4, c4 = (j & 15) << 2;
            *(float4*)&sl[(r << 6) + c4] =
                *(const float4*)&Xs[(size_t)(row0 + r) * 64 + c4];
        }
        for (int kb = 0; kb < 16; ++kb) {
            // A frag: 16x4 fp32; lanes 0-15 rows, hi half K+2  (ds_load_b64)
            v2f a = *(const v2f*)&sl[(lm << 6) + (kb << 2) + (half << 1)];
            const int kp = (hph << 5) + (kb << 1) + half;   // weight row-pair
            #pragma unroll
            for (int n = 0; n < 4; ++n) {
                v2f b = *(const v2f*)&wlds[((kp << 6) + (n << 4) + lm) * 2];
                acc[n] = __builtin_amdgcn_wmma_f32_16x16x4_f32(
                    false, a, false, b, (short)0, acc[n], false, false);
            }
        }
    }

    // bias + gelu, re-layout C-format -> row-major H in the wave slice
    #pragma unroll
    for (int n = 0; n < 4; ++n) {
        #pragma unroll
        for (int v = 0; v < 8; ++v) {
            // C layout: VGPR v -> row M = v + 8*half, col N = 16n + lm
            float h = acc[n][v] + B1[(n << 4) + lm];
            sl[((v + (half << 3)) << 6) + (n << 4) + lm] = gelu_tanh(h);
        }
    }

    // ---- GEMM2: O = H @ W2 + B2 -------------------------------------------
    v8f o[4];
    #pragma unroll
    for (int n = 0; n < 4; ++n) o[n] = (v8f)0.0f;

    for (int kb = 0; kb < 16; ++kb) {
        v2f a = *(const v2f*)&sl[(lm << 6) + (kb << 2) + (half << 1)];
        const int kp = (kb << 1) + half;
        #pragma unroll
        for (int n = 0; n < 4; ++n) {
            v2f b = *(const v2f*)&wlds[W2OFF + ((kp << 6) + (n << 4) + lm) * 2];
            o[n] = __builtin_amdgcn_wmma_f32_16x16x4_f32(
                false, a, false, b, (short)0, o[n], false, false);
        }
    }

    // stage O (+bias) into the slice (H fully consumed by GEMM2 above)
    #pragma unroll
    for (int n = 0; n < 4; ++n) {
        #pragma unroll
        for (int v = 0; v < 8; ++v) {
            sl[((v + (half << 3)) << 6) + (n << 4) + lm] = o[n][v] + B2[(n << 4) + lm];
        }
    }

    if (mode == 0) {
        if (active) {
            #pragma unroll
            for (int i = 0; i < 8; ++i) {
                int j = lane + (i << 5);
                int r = j >> 4, c4 = (j & 15) << 2;
                *(float4*)&Y[(size_t)(row0 + r) * 64 + c4] =
                    *(const float4*)&sl[(r << 6) + c4];
            }
        }
    } else {
        // l2-normalize each row, add residual
        float p = 0.f;
        #pragma unroll
        for (int c = 0; c < 32; ++c) {
            float v = sl[(lm << 6) + (half << 5) + c];
            p += v * v;
        }
        nr[(half << 4) + lm] = p;
        if (active) {
            #pragma unroll
            for (int i = 0; i < 8; ++i) {
                int j = lane + (i << 5);
                int r = j >> 4, c4 = (j & 15) << 2;
                float ss = nr[r] + nr[16 + r];
                float rs = rsqrtf(fmaxf(ss, 1e-12f));
                float4 vo = *(const float4*)&sl[(r << 6) + c4];
                const float4 rr = *(const float4*)&RES[(size_t)(row0 + r) * 64 + c4];
                float4 out;
                out.x = vo.x * rs + rr.x;
                out.y = vo.y * rs + rr.y;
                out.z = vo.z * rs + rr.z;
                out.w = vo.w * rs + rr.w;
                *(float4*)&Y[(size_t)(row0 + r) * 64 + c4] = out;
            }
        }
    }
}

// ---------------------------------------------------------------------------
// Edge gather-scale-scatter: agg[dst] += prep[src] * (ew[e] * inv_sum)
// 16 lanes per edge, float4 per lane; atomicAdd w/ unused return lowers to
// non-returning global_atomic_add_f32 (STOREcnt path).
// ---------------------------------------------------------------------------
__global__ __launch_bounds__(256)
void edge_scatter_kernel(const int* __restrict__ edges,
                         const float* __restrict__ ew,
                         const float* __restrict__ inv_sum,
                         const float* __restrict__ prep,
                         float* __restrict__ agg, int E)
{
    int t = blockIdx.x * blockDim.x + threadIdx.x;
    int e = t >> 4;
    if (e >= E) return;
    int c4 = (t & 15) * 4;
    int dst = edges[e];          // edges row 0: target node
    int src = edges[E + e];      // edges row 1: neighbour
    float w = ew[e] * inv_sum[0];
    const float4 m = *(const float4*)(prep + (size_t)src * 64 + c4);
    float* ap = agg + (size_t)dst * 64 + c4;
    atomicAdd(ap + 0, m.x * w);
    atomicAdd(ap + 1, m.y * w);
    atomicAdd(ap + 2, m.z * w);
    atomicAdd(ap + 3, m.w * w);
}

__global__ void zero4_kernel(float4* p, int n4) {
    int i = blockIdx.x * blockDim.x + threadIdx.x;
    if (i < n4) p[i] = make_float4(0.f, 0.f, 0.f, 0.f);
}

__global__ void reduce_partial_kernel(const float* __restrict__ w, float* partials, int n) {
    __shared__ float s[256];
    int i = blockIdx.x * 256 + threadIdx.x;
    s[threadIdx.x] = (i < n) ? w[i] : 0.0f;
    __syncthreads();
    for (int st = 128; st > 0; st >>= 1) {
        if ((int)threadIdx.x < st) s[threadIdx.x] += s[threadIdx.x + st];
        __syncthreads();
    }
    if (threadIdx.x == 0) partials[blockIdx.x] = s[0];
}

__global__ void reduce_final_kernel(const float* __restrict__ partials, int np, float* inv_out) {
    __shared__ float s[256];
    float acc = 0.0f;
    for (int i = threadIdx.x; i < np; i += 256) acc += partials[i];
    s[threadIdx.x] = acc;
    __syncthreads();
    for (int st = 128; st > 0; st >>= 1) {
        if ((int)threadIdx.x < st) s[threadIdx.x] += s[threadIdx.x + st];
        __syncthreads();
    }
    if (threadIdx.x == 0) inv_out[0] = 1.0f / s[0];
}

// logits: out[q,:] = X[idx[q]] @ W(64x2) + b
__global__ void logits_kernel(const float* __restrict__ X, const int* __restrict__ idx,
                              const float* __restrict__ W, const float* __restrict__ b,
                              float* __restrict__ out, int nq)
{
    int q = blockIdx.x * blockDim.x + threadIdx.x;
    if (q >= nq) return;
    const float* xr = X + (size_t)idx[q] * 64;
    float a0 = b[0], a1 = b[1];
    #pragma unroll 8
    for (int k = 0; k < 64; ++k) {
        float v = xr[k];
        a0 += v * W[k * 2];
        a1 += v * W[k * 2 + 1];
    }
    out[q * 2]     = a0;
    out[q * 2 + 1] = a1;
}

extern "C" void kernel_launch(void* const* d_in, const int* in_sizes, int n_in,
                              void* d_out, int out_size, void* d_ws, size_t ws_size,
                              hipStream_t stream) {
    const float* node_features = (const float*)d_in[0];
    const int*   edges         = (const int*)d_in[1];
    const float* edge_w        = (const float*)d_in[2];
    const int*   qidx          = (const int*)d_in[3];
    const float* pre_w1 = (const float*)d_in[4],  *pre_b1 = (const float*)d_in[5];
    const float* pre_w2 = (const float*)d_in[6],  *pre_b2 = (const float*)d_in[7];
    const float* c1p_w1 = (const float*)d_in[8],  *c1p_b1 = (const float*)d_in[9];
    const float* c1p_w2 = (const float*)d_in[10], *c1p_b2 = (const float*)d_in[11];
    const float* c1u_w1 = (const float*)d_in[12], *c1u_b1 = (const float*)d_in[13];
    const float* c1u_w2 = (const float*)d_in[14], *c1u_b2 = (const float*)d_in[15];
    const float* c2p_w1 = (const float*)d_in[16], *c2p_b1 = (const float*)d_in[17];
    const float* c2p_w2 = (const float*)d_in[18], *c2p_b2 = (const float*)d_in[19];
    const float* c2u_w1 = (const float*)d_in[20], *c2u_b1 = (const float*)d_in[21];
    const float* c2u_w2 = (const float*)d_in[22], *c2u_b2 = (const float*)d_in[23];
    const float* post_w1 = (const float*)d_in[24], *post_b1 = (const float*)d_in[25];
    const float* post_w2 = (const float*)d_in[26], *post_b2 = (const float*)d_in[27];
    const float* lg_w = (const float*)d_in[28], *lg_b = (const float*)d_in[29];

    const int N = GNN_N, E = GNN_E, NQ = GNN_Q;
    float* ws = (float*)d_ws;
    float* x        = ws;                      // N*64
    float* prep     = ws + (size_t)N * 64;     // N*64   (also reused for post)
    float* agg      = ws + (size_t)2 * N * 64; // N*64
    float* partials = ws + (size_t)3 * N * 64; // 4096
    float* inv_sum  = partials + 4096;         // 1

    const int NT = N / 16;                     // 3125 row tiles
    const int FB = (NT + 7) / 8;               // blocks (8 waves each)
    const int npart = (E + 255) / 256;         // 3125

    // 1. inv_sum = 1 / sum(edge_weights)  (deterministic two-pass)
    reduce_partial_kernel<<<npart, 256, 0, stream>>>(edge_w, partials, E);
    reduce_final_kernel<<<1, 256, 0, stream>>>(partials, npart, inv_sum);

    // 2. x = FFN_pre(node_features)
    ffn_tile_kernel<<<FB, 256, 0, stream>>>(node_features, nullptr,
        pre_w1, pre_b1, pre_w2, pre_b2, nullptr, x, NT, 64, 0);

    // 3. conv1: prepare once per NODE (FFN is row-wise => FFN(x[nbr])=FFN(x)[nbr])
    ffn_tile_kernel<<<FB, 256, 0, stream>>>(x, nullptr,
        c1p_w1, c1p_b1, c1p_w2, c1p_b2, nullptr, prep, NT, 64, 0);
    zero4_kernel<<<(N * 16 + 255) / 256, 256, 0, stream>>>((float4*)agg, N * 16);
    edge_scatter_kernel<<<(E * 16 + 255) / 256, 256, 0, stream>>>(
        edges, edge_w, inv_sum, prep, agg, E);
    //   update: l2norm(FFN([x,agg])) + x   (in-place safe: each wave owns its rows)
    ffn_tile_kernel<<<FB, 256, 0, stream>>>(x, agg,
        c1u_w1, c1u_b1, c1u_w2, c1u_b2, x, x, NT, 128, 1);

    // 4. conv2
    ffn_tile_kernel<<<FB, 256, 0, stream>>>(x, nullptr,
        c2p_w1, c2p_b1, c2p_w2, c2p_b2, nullptr, prep, NT, 64, 0);
    zero4_kernel<<<(N * 16 + 255) / 256, 256, 0, stream>>>((float4*)agg, N * 16);
    edge_scatter_kernel<<<(E * 16 + 255) / 256, 256, 0, stream>>>(
        edges, edge_w, inv_sum, prep, agg, E);
    ffn_tile_kernel<<<FB, 256, 0, stream>>>(x, agg,
        c2u_w1, c2u_b1, c2u_w2, c2u_b2, x, x, NT, 128, 1);

    // 5. post FFN (into prep buffer)
    ffn_tile_kernel<<<FB, 256, 0, stream>>>(x, nullptr,
        post_w1, post_b1, post_w2, post_b2, nullptr, prep, NT, 64, 0);

    // 6. gather queries + logits
    logits_kernel<<<(NQ + 255) / 256, 256, 0, stream>>>(
        prep, qidx, lg_w, lg_b, (float*)d_out, NQ);
}